// ConvNextMaskRCNNRPN_8873402434043
// MI455X (gfx1250) — compile-verified
//
#include <hip/hip_runtime.h>
#include <math.h>

#define BATCH    16
#define NANCH    300000
#define NMSPRE   2000
#define KPAD     2048
#define MAXPER   1000
#define IOU_THR  0.7f
#define IMGW     1024.0f
#define IMGH     1024.0f
#define MAXRATIO 4.135166556742356f   /* |log(16/1000)| */
#define NWORDS   64                   /* 2048 bits / 32 */

typedef __attribute__((ext_vector_type(2))) float v2f;
typedef __attribute__((ext_vector_type(8))) float v8f;

__device__ __forceinline__ unsigned f2key(float f) {
  unsigned u = __float_as_uint(f);
  return (u & 0x80000000u) ? ~u : (u | 0x80000000u);
}

// ---------------------------------------------------------------------------
// K1: per-image radix-select top-2000 + bitonic sort (desc) in LDS.
// One 1024-thread workgroup per image. Pads slots 2000..2047 with -inf.
// ---------------------------------------------------------------------------
__global__ __launch_bounds__(1024)
void k_select_sort(const float* __restrict__ scores,
                   int* __restrict__ sel_idx, float* __restrict__ sel_score) {
  const int img = blockIdx.x;
  const int tid = threadIdx.x;
  const float* s = scores + (size_t)img * NANCH;

  __shared__ unsigned hist[256];
  __shared__ unsigned s_prefix;
  __shared__ int s_want;
  __shared__ int s_ctr_gt, s_ctr_eq;
  __shared__ float ls[KPAD];
  __shared__ int   lidx[KPAD];

  if (tid == 0) { s_prefix = 0u; s_want = NMSPRE; }

  // 4-pass MSB radix select of the 2000th-largest key
  for (int pass = 0; pass < 4; ++pass) {
    const int shift = 24 - pass * 8;
    for (int i = tid; i < 256; i += 1024) hist[i] = 0u;
    __syncthreads();
    const unsigned pref = s_prefix;
    for (int i = tid; i < NANCH; i += 1024) {
      unsigned key = f2key(s[i]);
      bool match = (pass == 0) || ((key >> (shift + 8)) == (pref >> (shift + 8)));
      if (match) atomicAdd(&hist[(key >> shift) & 255u], 1u);
    }
    __syncthreads();
    if (tid == 0) {
      int w = s_want;
      int d = 255;
      while (d > 0 && (int)hist[d] < w) { w -= (int)hist[d]; --d; }
      s_prefix = pref | ((unsigned)d << shift);
      s_want = w;
    }
    __syncthreads();
  }

  const unsigned T = s_prefix;       // threshold key
  const int nties = s_want;          // take this many keys == T
  const int ngt   = NMSPRE - nties;  // number of keys strictly > T

  if (tid == 0) { s_ctr_gt = 0; s_ctr_eq = 0; }
  for (int i = tid; i < KPAD; i += 1024) { ls[i] = -INFINITY; lidx[i] = 0; }
  __syncthreads();

  for (int i = tid; i < NANCH; i += 1024) {
    float v = s[i];
    unsigned key = f2key(v);
    if (key > T) {
      int p = atomicAdd(&s_ctr_gt, 1);
      ls[p] = v; lidx[p] = i;
    } else if (key == T) {
      int q = atomicAdd(&s_ctr_eq, 1);
      if (q < nties) { ls[ngt + q] = v; lidx[ngt + q] = i; }
    }
  }
  __syncthreads();

  // bitonic sort 2048 entries, descending by score
  for (unsigned k2 = 2; k2 <= KPAD; k2 <<= 1) {
    for (unsigned jj = k2 >> 1; jj > 0; jj >>= 1) {
      __syncthreads();
      for (unsigned i = tid; i < KPAD; i += 1024) {
        unsigned ix = i ^ jj;
        if (ix > i) {
          bool desc = ((i & k2) == 0u);
          float a = ls[i], b = ls[ix];
          bool sw = desc ? (a < b) : (a > b);
          if (sw) {
            ls[i] = b; ls[ix] = a;
            int t = lidx[i]; lidx[i] = lidx[ix]; lidx[ix] = t;
          }
        }
      }
    }
  }
  __syncthreads();

  for (int i = tid; i < KPAD; i += 1024) {
    sel_score[img * KPAD + i] = ls[i];
    sel_idx  [img * KPAD + i] = lidx[i];
  }
}

// ---------------------------------------------------------------------------
// K2: gather + delta2bbox decode + clamp; per-image max coord; level offsets.
// ---------------------------------------------------------------------------
__global__ __launch_bounds__(1024)
void k_decode(const float* __restrict__ anchors, const float* __restrict__ deltas,
              const int* __restrict__ level_ids, const int* __restrict__ sel_idx,
              const float* __restrict__ sel_score,
              float* __restrict__ boxes, float* __restrict__ boxes_nms) {
  const int img = blockIdx.x;
  const int tid = threadIdx.x;
  __shared__ float red[1024];

  float bx[2][4]; int ids[2]; bool real[2];
  float mymax = 0.0f;

  #pragma unroll
  for (int e = 0; e < 2; ++e) {
    int k = tid + e * 1024;
    float sc = sel_score[img * KPAD + k];
    int id = sel_idx[img * KPAD + k];
    ids[e] = id;
    bool isreal = (sc > -1e37f);
    real[e] = isreal;
    float x1 = 0.f, y1 = 0.f, x2 = 0.f, y2 = 0.f;
    if (isreal) {
      const float* a = anchors + ((size_t)img * NANCH + id) * 4;
      const float* d = deltas  + ((size_t)img * NANCH + id) * 4;
      float a0 = a[0], a1 = a[1], a2 = a[2], a3 = a[3];
      float dx = d[0], dy = d[1];
      float dw = fminf(fmaxf(d[2], -MAXRATIO), MAXRATIO);
      float dh = fminf(fmaxf(d[3], -MAXRATIO), MAXRATIO);
      float px = (a0 + a2) * 0.5f, py = (a1 + a3) * 0.5f;
      float pw = a2 - a0,          ph = a3 - a1;
      float gx = px + pw * dx,     gy = py + ph * dy;
      float gw = pw * __expf(dw),  gh = ph * __expf(dh);
      x1 = fminf(fmaxf(gx - gw * 0.5f, 0.f), IMGW);
      y1 = fminf(fmaxf(gy - gh * 0.5f, 0.f), IMGH);
      x2 = fminf(fmaxf(gx + gw * 0.5f, 0.f), IMGW);
      y2 = fminf(fmaxf(gy + gh * 0.5f, 0.f), IMGH);
      mymax = fmaxf(mymax, fmaxf(fmaxf(x1, y1), fmaxf(x2, y2)));
    }
    bx[e][0] = x1; bx[e][1] = y1; bx[e][2] = x2; bx[e][3] = y2;
    float* o = boxes + ((size_t)img * KPAD + k) * 4;
    o[0] = x1; o[1] = y1; o[2] = x2; o[3] = y2;
  }

  red[tid] = mymax;
  __syncthreads();
  for (int off = 512; off > 0; off >>= 1) {
    if (tid < off) red[tid] = fmaxf(red[tid], red[tid + off]);
    __syncthreads();
  }
  const float maxc = red[0];

  #pragma unroll
  for (int e = 0; e < 2; ++e) {
    int k = tid + e * 1024;
    float off = 0.f;
    if (real[e])
      off = (float)level_ids[(size_t)img * NANCH + ids[e]] * (maxc + 1.0f);
    float* o = boxes_nms + ((size_t)img * KPAD + k) * 4;
    o[0] = bx[e][0] + off; o[1] = bx[e][1] + off;
    o[2] = bx[e][2] + off; o[3] = bx[e][3] + off;
  }
}

// ---------------------------------------------------------------------------
// K3: suppression bit-matrix. One wave per 16-row x 32-col tile.
// WMMA f32 16x16x4 computes the union base term area_i + area_j per tile.
// Division-free threshold: inter/max(u,eps) > THR  <=>  inter > THR*max(u,eps)
// ---------------------------------------------------------------------------
__global__ __launch_bounds__(32)
void k_iou_mask(const float* __restrict__ boxes_nms, unsigned* __restrict__ mask) {
  const int colT = blockIdx.x;   // 0..63  (32 columns each)
  const int rowT = blockIdx.y;   // 0..127 (16 rows each)
  const int img  = blockIdx.z;
  const int lane = threadIdx.x;  // wave32

  __shared__ float rb[16][4];
  __shared__ float cb[32][4];
  const float* bb = boxes_nms + (size_t)img * KPAD * 4;

  if (lane < 16) {
    const float* p = bb + (size_t)(rowT * 16 + lane) * 4;
    rb[lane][0] = p[0]; rb[lane][1] = p[1]; rb[lane][2] = p[2]; rb[lane][3] = p[3];
  }
  {
    const float* p = bb + (size_t)(colT * 32 + lane) * 4;
    cb[lane][0] = p[0]; cb[lane][1] = p[1]; cb[lane][2] = p[2]; cb[lane][3] = p[3];
  }
  __syncthreads();

  // A (16x4): rows = [area_row, 1, 0, 0]; B (4x16): cols = [1, area_col, 0, 0]^T
  // -> C[i][j] = area_i + area_j. Lanes 0-15 carry K=0,1; lanes 16-31 carry K=2,3 (zero).
  v2f a, b0, b1;
  if (lane < 16) {
    float ar  = (rb[lane][2] - rb[lane][0]) * (rb[lane][3] - rb[lane][1]);
    a.x = ar; a.y = 1.0f;
    float ac0 = (cb[lane][2] - cb[lane][0]) * (cb[lane][3] - cb[lane][1]);
    b0.x = 1.0f; b0.y = ac0;
    int n1 = lane + 16;
    float ac1 = (cb[n1][2] - cb[n1][0]) * (cb[n1][3] - cb[n1][1]);
    b1.x = 1.0f; b1.y = ac1;
  } else {
    a.x = 0.f; a.y = 0.f; b0.x = 0.f; b0.y = 0.f; b1.x = 0.f; b1.y = 0.f;
  }
  v8f z = {0.f, 0.f, 0.f, 0.f, 0.f, 0.f, 0.f, 0.f};
  v8f c0 = __builtin_amdgcn_wmma_f32_16x16x4_f32(false, a, false, b0, (short)0, z, false, false);
  v8f c1 = __builtin_amdgcn_wmma_f32_16x16x4_f32(false, a, false, b1, (short)0, z, false, false);

  const int lr = (lane < 16) ? 0 : 8;   // C layout: VGPR r holds rows r (lanes 0-15), r+8 (lanes 16-31)
  const int n  = lane & 15;
  const float c0x1 = cb[n][0],      c0y1 = cb[n][1],      c0x2 = cb[n][2],      c0y2 = cb[n][3];
  const float c1x1 = cb[n + 16][0], c1y1 = cb[n + 16][1], c1x2 = cb[n + 16][2], c1y2 = cb[n + 16][3];
  const unsigned rowbase = ((unsigned)img * KPAD + (unsigned)rowT * 16) * NWORDS;

  #pragma unroll
  for (int r = 0; r < 8; ++r) {
    const int lrow = r + lr;
    const int grow = rowT * 16 + lrow;
    const float rx1 = rb[lrow][0], ry1 = rb[lrow][1], rx2 = rb[lrow][2], ry2 = rb[lrow][3];

    float iw0 = fminf(rx2, c0x2) - fmaxf(rx1, c0x1);
    float ih0 = fminf(ry2, c0y2) - fmaxf(ry1, c0y1);
    float inter0 = fmaxf(iw0, 0.f) * fmaxf(ih0, 0.f);
    float u0 = c0[r] - inter0;
    int gj0 = colT * 32 + n;
    bool p0 = (inter0 > IOU_THR * fmaxf(u0, 1e-6f)) && (gj0 > grow);
    unsigned B0 = (unsigned)__ballot(p0);

    float iw1 = fminf(rx2, c1x2) - fmaxf(rx1, c1x1);
    float ih1 = fminf(ry2, c1y2) - fmaxf(ry1, c1y1);
    float inter1 = fmaxf(iw1, 0.f) * fmaxf(ih1, 0.f);
    float u1 = c1[r] - inter1;
    int gj1 = colT * 32 + 16 + n;
    bool p1 = (inter1 > IOU_THR * fmaxf(u1, 1e-6f)) && (gj1 > grow);
    unsigned B1 = (unsigned)__ballot(p1);

    unsigned w_lo = (B0 & 0xFFFFu) | ((B1 & 0xFFFFu) << 16);   // row r,   cols 0..31
    unsigned w_hi = (B0 >> 16)     | (B1 & 0xFFFF0000u);       // row r+8, cols 0..31
    if (lane == 0) mask[rowbase + (unsigned)r       * NWORDS + colT] = w_lo;
    if (lane == 1) mask[rowbase + (unsigned)(r + 8) * NWORDS + colT] = w_hi;
  }
}

// ---------------------------------------------------------------------------
// K4: greedy NMS scan, one wave per image. The 512 KB/img suppression matrix
// is streamed ahead of the scan pointer into an 8 KB LDS ring with
// global_load_async_to_lds_b128 (ASYNCcnt), so the 2048-step serial scan only
// ever touches LDS. 1 group = 2 rows = 512 B = one async instruction
// (32 lanes x 16 B). Ring = 16 groups; async loads complete in order, so the
// steady-state wait is the constant s_wait_asynccnt 15.
// ---------------------------------------------------------------------------
#define RINGG 16                        // groups in ring (2 rows each)

__device__ __forceinline__ void async_issue_group(const unsigned* m, unsigned ring_off,
                                                  int g, int lane) {
  unsigned long long ga = (unsigned long long)(uintptr_t)m +
                          (unsigned long long)g * 512ull + (unsigned)(lane * 16);
  unsigned la = ring_off + (unsigned)(g & (RINGG - 1)) * 512u + (unsigned)(lane * 16);
  asm volatile("global_load_async_to_lds_b128 %0, %1, off"
               :: "v"(la), "v"(ga) : "memory");
}

__device__ __forceinline__ void nms_apply_row(volatile unsigned* rem,
                                              const unsigned* ringrow,
                                              int i, int lane) {
  unsigned w = rem[i >> 5];                       // uniform broadcast read
  if (((w >> (i & 31)) & 1u) == 0u) {            // row i survives: apply suppression
    unsigned r0 = ringrow[lane * 2];
    unsigned r1 = ringrow[lane * 2 + 1];
    rem[lane * 2]     |= r0;
    rem[lane * 2 + 1] |= r1;
  }
}

__global__ __launch_bounds__(32)
void k_nms_scan(const unsigned* __restrict__ mask, unsigned* __restrict__ keepw) {
  const int img  = blockIdx.x;
  const int lane = threadIdx.x;
  volatile __shared__ unsigned rem[NWORDS];
  __shared__ unsigned ring[RINGG * 128];          // 16 groups x 512 B = 8 KB
  rem[lane] = 0u; rem[lane + 32] = 0u;
  __syncthreads();

  const unsigned* m = mask + (size_t)img * KPAD * NWORDS;
  const unsigned ring_off = (unsigned)(uintptr_t)&ring[0];
  const int G = KPAD / 2;                         // 1024 groups of 2 rows

  // prologue: fill the ring
  for (int g = 0; g < RINGG; ++g) async_issue_group(m, ring_off, g, lane);

  // steady state: wait for oldest group, consume it, refill
  for (int g = 0; g < G - RINGG; ++g) {
    asm volatile("s_wait_asynccnt 0xf" ::: "memory");
    const unsigned* slot = &ring[(g & (RINGG - 1)) * 128];
    nms_apply_row(rem, slot,      2 * g,     lane);
    nms_apply_row(rem, slot + 64, 2 * g + 1, lane);
    async_issue_group(m, ring_off, g + RINGG, lane);
  }
  // drain: everything outstanding has been issued; wait once, consume tail
  asm volatile("s_wait_asynccnt 0x0" ::: "memory");
  for (int g = G - RINGG; g < G; ++g) {
    const unsigned* slot = &ring[(g & (RINGG - 1)) * 128];
    nms_apply_row(rem, slot,      2 * g,     lane);
    nms_apply_row(rem, slot + 64, 2 * g + 1, lane);
  }

  __syncthreads();
  keepw[img * NWORDS + lane]      = ~rem[lane];
  keepw[img * NWORDS + lane + 32] = ~rem[lane + 32];
}

// ---------------------------------------------------------------------------
// K5: scores were pre-sorted desc, so top-1000 of kept == stable compaction.
// ---------------------------------------------------------------------------
__global__ __launch_bounds__(256)
void k_finalize(const unsigned* __restrict__ keepw, const float* __restrict__ sel_score,
                const float* __restrict__ boxes, float* __restrict__ out) {
  const int img = blockIdx.x;
  const int tid = threadIdx.x;
  __shared__ unsigned kw[NWORDS];
  __shared__ int csum[256];
  __shared__ int s_total;

  if (tid < NWORDS) kw[tid] = keepw[img * NWORDS + tid];
  __syncthreads();

  const int base = tid * 8;
  bool kp[8]; float sc[8];
  int cnt = 0;
  #pragma unroll
  for (int e = 0; e < 8; ++e) {
    int k = base + e;
    float v = sel_score[img * KPAD + k];
    bool b = (((kw[k >> 5] >> (k & 31)) & 1u) != 0u) && (v > -1e37f);
    kp[e] = b; sc[e] = v; cnt += b ? 1 : 0;
  }
  csum[tid] = cnt;
  __syncthreads();
  for (int off = 1; off < 256; off <<= 1) {
    int v = (tid >= off) ? csum[tid - off] : 0;
    __syncthreads();
    csum[tid] += v;
    __syncthreads();
  }
  if (tid == 255) s_total = csum[255];
  int p = csum[tid] - cnt;   // exclusive prefix of kept count
  __syncthreads();

  #pragma unroll
  for (int e = 0; e < 8; ++e) {
    if (kp[e]) {
      if (p < MAXPER) {
        const float* bp = boxes + ((size_t)img * KPAD + base + e) * 4;
        float* o = out + ((size_t)img * MAXPER + p) * 5;
        o[0] = bp[0]; o[1] = bp[1]; o[2] = bp[2]; o[3] = bp[3]; o[4] = sc[e];
      }
      ++p;
    }
  }
  const int total = s_total;
  for (int q = tid; q < MAXPER; q += 256) {
    if (q >= total) {
      float* o = out + ((size_t)img * MAXPER + q) * 5;
      o[0] = 0.f; o[1] = 0.f; o[2] = 0.f; o[3] = 0.f; o[4] = 0.f;
    }
  }
}

// ---------------------------------------------------------------------------
extern "C" void kernel_launch(void* const* d_in, const int* in_sizes, int n_in,
                              void* d_out, int out_size, void* d_ws, size_t ws_size,
                              hipStream_t stream) {
  (void)in_sizes; (void)n_in; (void)out_size; (void)ws_size;
  const float* anchors = (const float*)d_in[0];
  const float* deltas  = (const float*)d_in[1];
  const float* scores  = (const float*)d_in[2];
  const int*   levels  = (const int*)d_in[3];

  char* ws = (char*)d_ws;
  size_t off = 0;
  auto alloc = [&](size_t bytes) -> void* {
    void* p = ws + off;
    off = (off + bytes + 255) & ~(size_t)255;
    return p;
  };
  int*      sel_idx   = (int*)     alloc((size_t)BATCH * KPAD * 4);
  float*    sel_score = (float*)   alloc((size_t)BATCH * KPAD * 4);
  float*    boxes     = (float*)   alloc((size_t)BATCH * KPAD * 16);
  float*    boxes_nms = (float*)   alloc((size_t)BATCH * KPAD * 16);
  unsigned* keepw     = (unsigned*)alloc((size_t)BATCH * NWORDS * 4);
  unsigned* mask      = (unsigned*)alloc((size_t)BATCH * KPAD * NWORDS * 4); // 8 MB

  k_select_sort<<<BATCH, 1024, 0, stream>>>(scores, sel_idx, sel_score);
  k_decode<<<BATCH, 1024, 0, stream>>>(anchors, deltas, levels, sel_idx, sel_score,
                                       boxes, boxes_nms);
  k_iou_mask<<<dim3(NWORDS, KPAD / 16, BATCH), 32, 0, stream>>>(boxes_nms, mask);
  k_nms_scan<<<BATCH, 32, 0, stream>>>(mask, keepw);
  k_finalize<<<BATCH, 256, 0, stream>>>(keepw, sel_score, boxes, (float*)d_out);
}